// FourierBlock_15728170238242
// MI455X (gfx1250) — compile-verified
//
#include <hip/hip_runtime.h>
#include <hip/hip_bf16.h>

// Problem constants (from reference)
constexpr int Bn = 16;      // batch
constexpr int Ln = 4096;    // sequence length
constexpr int Hn = 3;       // heads
constexpr int En = 512;     // embed per head
constexpr int Mn = 128;     // selected frequency modes
constexpr int HE = Hn * En;           // 1536
constexpr int Rn = Bn * Hn * En;      // 24576 rows (b,h,e)

typedef float v2f __attribute__((ext_vector_type(2)));
typedef float v8f __attribute__((ext_vector_type(8)));

static __device__ __forceinline__ v8f wmma_f32(v2f a, v2f b, v8f c) {
    // 8 args: (neg_a, A, neg_b, B, c_mod, C, reuse_a, reuse_b)
    return __builtin_amdgcn_wmma_f32_16x16x4_f32(false, a, false, b, (short)0, c,
                                                 false, false);
}

// --- CDNA5 async global->LDS copies (ASYNCcnt-tracked) ----------------------
static __device__ __forceinline__ void async_copy_b128(uint32_t lds_off,
                                                       const void* gptr) {
    asm volatile("global_load_async_to_lds_b128 %0, %1, off"
                 :: "v"(lds_off), "v"(gptr) : "memory");
}
static __device__ __forceinline__ void async_copy_b32(uint32_t lds_off,
                                                      const void* gptr) {
    asm volatile("global_load_async_to_lds_b32 %0, %1, off"
                 :: "v"(lds_off), "v"(gptr) : "memory");
}
#define S_WAIT_ASYNC_2() asm volatile("s_wait_asynccnt 0x2" ::: "memory")
#define S_WAIT_ASYNC_0() asm volatile("s_wait_asynccnt 0x0" ::: "memory")

static __device__ __forceinline__ uint32_t lds_addr(const void* p) {
    // generic pointer into LDS aperture: low 32 bits are the LDS offset
    return (uint32_t)(uintptr_t)p;
}

// ---------------------------------------------------------------------------
// K0: build twiddle tables from the selected mode indices.
//   twF : [L][2M]  twF[l][m]    =  cos(2*pi*k_m*l/L)
//                  twF[l][M+m]  = -sin(2*pi*k_m*l/L)          (forward DFT)
//   twI : [2M][L]  twI[m][l]    =  f_m*cos(.)/L
//                  twI[M+m][l]  = -f_m*sin(.)/L,  f_m = (k_m==0 ? 1 : 2)
// ---------------------------------------------------------------------------
__global__ __launch_bounds__(256) void fb_twiddles(const int* __restrict__ index,
                                                   float* __restrict__ twF,
                                                   float* __restrict__ twI) {
    int tid = blockIdx.x * blockDim.x + threadIdx.x;   // over M*L
    if (tid >= Mn * Ln) return;
    int m = tid >> 12;          // / 4096
    int l = tid & (Ln - 1);
    int k = index[m];
    // phase reduced mod L: k*l < 2^24 so exact in int; keeps arg small for sincos
    int ph = (int)(((long long)k * (long long)l) & (long long)(Ln - 1));
    float theta = (float)ph * (6.283185307179586f / (float)Ln);
    float s, c;
    __sincosf(theta, &s, &c);
    twF[(size_t)l * (2 * Mn) + m]       = c;
    twF[(size_t)l * (2 * Mn) + Mn + m]  = -s;
    float f = ((k == 0) ? 1.0f : 2.0f) / (float)Ln;
    twI[(size_t)m * Ln + l]             = f * c;
    twI[(size_t)(Mn + m) * Ln + l]      = -f * s;
}

// ---------------------------------------------------------------------------
// K1: forward sparse DFT, 32 rows x 128 modes per block.
//   X[b,h,e,m] (re,im) = sum_l q[b,l,h,e] * twiddle
//   The 64x32 q chunk is staged per block via async b128 copies into two
//   16-wide LDS panels (conflict-free stride-16), double buffered; each
//   streamed twiddle B-fragment then feeds 4 WMMAs (2 panels x re/im).
//   Output layout transposed for stage 2:  Xr2/Xi2 [h][m][b][i(=e)]
// ---------------------------------------------------------------------------
__global__ __launch_bounds__(256) void fb_dft_fwd(const float* __restrict__ q,
                                                  const float* __restrict__ twF,
                                                  float* __restrict__ Xr2,
                                                  float* __restrict__ Xi2) {
    __shared__ float xs[2][2][64 * 16];            // [buf][panel][l*16+e], 16 KB

    const int lane = threadIdx.x & 31;
    const int wave = threadIdx.x >> 5;
    const int row0 = blockIdx.x * 32;          // (b,h,e) flat row tile (32 wide)
    const int b    = row0 / HE;
    const int rem  = row0 % HE;
    const int h    = rem / En;
    const int e0   = rem % En;
    const int m0   = wave * 16;

    const int rn = lane & 15;                  // tile row (A) / tile col (B,D)
    const int kb = (lane >> 4) << 1;           // 0 or 2
    const int hi = lane >> 4;

    // q[b][l][h][e] : element (l, e) at qblk[l*HE + e]
    const float* qblk = q + (size_t)b * Ln * HE + (size_t)h * En + e0;

    // staging role: two b128 (panel 0: e, panel 1: e+16) per chunk
    const int sl = threadIdx.x >> 2;           // 0..63 : l within chunk
    const int se = (threadIdx.x & 3) * 4;      // 0,4,8,12

    constexpr int KC = 64;                     // l-chunk
    constexpr int NC = Ln / KC;                // 64 chunks

    {   // issue chunk 0
        const float* src = qblk + (size_t)sl * HE + se;
        async_copy_b128(lds_addr(&xs[0][0][sl * 16 + se]), src);
        async_copy_b128(lds_addr(&xs[0][1][sl * 16 + se]), src + 16);
    }

    v8f aR0 = {}, aI0 = {}, aR1 = {}, aI1 = {};
    for (int c = 0; c < NC; ++c) {
        if (c + 1 < NC) {
            const float* src = qblk + (size_t)((c + 1) * KC + sl) * HE + se;
            const int nb = (c + 1) & 1;
            async_copy_b128(lds_addr(&xs[nb][0][sl * 16 + se]), src);
            async_copy_b128(lds_addr(&xs[nb][1][sl * 16 + se]), src + 16);
            S_WAIT_ASYNC_2();                  // chunk c's 2 copies have landed
        } else {
            S_WAIT_ASYNC_0();
        }
        __syncthreads();

        const int buf = c & 1;
        const int l0c = c * KC;
        for (int lc = 0; lc < KC; lc += 4) {
            v2f a0, a1;
            a0.x = xs[buf][0][(lc + kb) * 16 + rn];
            a0.y = xs[buf][0][(lc + kb + 1) * 16 + rn];
            a1.x = xs[buf][1][(lc + kb) * 16 + rn];
            a1.y = xs[buf][1][(lc + kb + 1) * 16 + rn];
            const float* t0 = twF + (size_t)(l0c + lc + kb) * (2 * Mn) + m0 + rn;
            const float* t1 = t0 + 2 * Mn;
            v2f bc; bc.x = t0[0];  bc.y = t1[0];
            v2f bs; bs.x = t0[Mn]; bs.y = t1[Mn];
            aR0 = wmma_f32(a0, bc, aR0);
            aI0 = wmma_f32(a0, bs, aI0);
            aR1 = wmma_f32(a1, bc, aR1);
            aI1 = wmma_f32(a1, bs, aI1);
        }
        __syncthreads();                       // done reading buf before reuse
    }

    // D tile: VGPR v -> tile row v (lanes 0-15) / v+8 (lanes 16-31); col = rn
    const size_t base = ((size_t)(h * Mn + m0 + rn) * Bn + b) * En + e0;
#pragma unroll
    for (int v = 0; v < 8; ++v) {
        int er = v + hi * 8;
        Xr2[base + er]      = aR0[v];
        Xi2[base + er]      = aI0[v];
        Xr2[base + 16 + er] = aR1[v];
        Xi2[base + 16 + er] = aI1[v];
    }
}

// ---------------------------------------------------------------------------
// K2: per-(h,m) complex channel mixing.
//   O[b,h,o,m] = sum_i X[b,h,i,m] * W[h,i,o,m]      (complex)
//   All 8 waves of a block share one (h,m): the 16x512 X panels are staged
//   once into LDS transposed to [i][b] (async b32 scatter), each wave owns
//   2 o-tiles so every A fragment feeds 8 WMMAs.
//   Or/Oi layout: [b][h][o][m]  (rows (b,h,o) x M, m contiguous -> stage 3 A)
// ---------------------------------------------------------------------------
__global__ __launch_bounds__(256) void fb_modemix(const float* __restrict__ Xr2,
                                                  const float* __restrict__ Xi2,
                                                  const float* __restrict__ wr,
                                                  const float* __restrict__ wi,
                                                  float* __restrict__ Or,
                                                  float* __restrict__ Oi) {
    __shared__ float XsR[En * 16];                 // [i][b], 32 KB
    __shared__ float XsI[En * 16];                 // [i][b], 32 KB

    const int lane = threadIdx.x & 31;
    const int wave = threadIdx.x >> 5;
    const int hm   = blockIdx.x >> 1;              // 0..383
    const int og   = blockIdx.x & 1;               // which half of o range
    const int h    = hm / Mn;
    const int m    = hm % Mn;
    const int o0a  = (og * 16 + wave) * 16;        // first o-tile
    const int o0b  = o0a + 128;                    // second o-tile (+8 tiles)

    const int rn = lane & 15;
    const int kb = (lane >> 4) << 1;
    const int hi = lane >> 4;

    // X slices: [b][i] with element xr[b*En + i]
    const float* xr = Xr2 + (size_t)(h * Mn + m) * Bn * En;
    const float* xi = Xi2 + (size_t)(h * Mn + m) * Bn * En;

    // stage X transposed into LDS: thread (bb, ig) scatters i = ig + 16*j
    {
        const int bb = threadIdx.x >> 4;           // 0..15 batch row
        const int ig = threadIdx.x & 15;
#pragma unroll
        for (int j = 0; j < 32; ++j) {
            int i = ig + 16 * j;
            async_copy_b32(lds_addr(&XsR[i * 16 + bb]), xr + (size_t)bb * En + i);
            async_copy_b32(lds_addr(&XsI[i * 16 + bb]), xi + (size_t)bb * En + i);
        }
        S_WAIT_ASYNC_0();
        __syncthreads();
    }

    // W: [h][i][o][m]; B element (k=i, n=o): wrb[(i0+k) * En*Mn]
    const size_t woff = (size_t)h * En * En * Mn + (size_t)(o0a + rn) * Mn + m;
    const float* wrb0 = wr + woff;
    const float* wib0 = wi + woff;
    const float* wrb1 = wrb0 + (size_t)128 * Mn;   // o + 128
    const float* wib1 = wib0 + (size_t)128 * Mn;
    const size_t istride = (size_t)En * Mn;

    v8f rr0 = {}, ii0 = {}, ri0 = {}, ir0 = {};
    v8f rr1 = {}, ii1 = {}, ri1 = {}, ir1 = {};
    for (int i0 = 0; i0 < En; i0 += 4) {
        v2f ar, ai;
        ar.x = XsR[(i0 + kb) * 16 + rn];
        ar.y = XsR[(i0 + kb + 1) * 16 + rn];
        ai.x = XsI[(i0 + kb) * 16 + rn];
        ai.y = XsI[(i0 + kb + 1) * 16 + rn];
        v2f br0, bi0, br1, bi1;
        br0.x = wrb0[(size_t)(i0 + kb) * istride];
        br0.y = wrb0[(size_t)(i0 + kb + 1) * istride];
        bi0.x = wib0[(size_t)(i0 + kb) * istride];
        bi0.y = wib0[(size_t)(i0 + kb + 1) * istride];
        br1.x = wrb1[(size_t)(i0 + kb) * istride];
        br1.y = wrb1[(size_t)(i0 + kb + 1) * istride];
        bi1.x = wib1[(size_t)(i0 + kb) * istride];
        bi1.y = wib1[(size_t)(i0 + kb + 1) * istride];
        rr0 = wmma_f32(ar, br0, rr0);
        ii0 = wmma_f32(ai, bi0, ii0);
        ri0 = wmma_f32(ar, bi0, ri0);
        ir0 = wmma_f32(ai, br0, ir0);
        rr1 = wmma_f32(ar, br1, rr1);
        ii1 = wmma_f32(ai, bi1, ii1);
        ri1 = wmma_f32(ar, bi1, ri1);
        ir1 = wmma_f32(ai, br1, ir1);
    }

    // combine complex parts and store: row b = v + hi*8, col o = o0 + rn
#pragma unroll
    for (int v = 0; v < 8; ++v) {
        int bb = v + hi * 8;
        size_t idxa = ((size_t)(bb * Hn + h) * En + (o0a + rn)) * Mn + m;
        size_t idxb = ((size_t)(bb * Hn + h) * En + (o0b + rn)) * Mn + m;
        Or[idxa] = rr0[v] - ii0[v];
        Oi[idxa] = ri0[v] + ir0[v];
        Or[idxb] = rr1[v] - ii1[v];
        Oi[idxb] = ri1[v] + ir1[v];
    }
}

// ---------------------------------------------------------------------------
// K3: inverse sparse DFT (irfft of the M-sparse spectrum), scale folded in twI.
//   out[row][l] = sum_{k<2M} A[row][k] * twI[k][l]
//   32 rows per block: the 32x256 [Or|Oi] panel is scattered once into two
//   16-row LDS panels (transposed to [k][r]); each streamed twI B-fragment
//   feeds 2 WMMAs.
// ---------------------------------------------------------------------------
__global__ __launch_bounds__(256) void fb_idft(const float* __restrict__ Or,
                                               const float* __restrict__ Oi,
                                               const float* __restrict__ twI,
                                               float* __restrict__ out) {
    __shared__ float As[2][2 * Mn * 16];           // [panel][k*16+r], 32 KB

    const int lane = threadIdx.x & 31;
    const int row0 = blockIdx.x * 32;
    const int l0   = blockIdx.y * 128 + (threadIdx.x >> 5) * 16;

    const int rn = lane & 15;
    const int kb = (lane >> 4) << 1;
    const int hi = lane >> 4;

    // cooperative transposed staging: thread (r32, tm) moves m = tm + 8*j
    {
        const int r32   = threadIdx.x >> 3;        // 0..31 panel row
        const int tm    = threadIdx.x & 7;
        const int panel = r32 >> 4;
        const int rr    = r32 & 15;
        const float* orow = Or + (size_t)(row0 + r32) * Mn;
        const float* irow = Oi + (size_t)(row0 + r32) * Mn;
#pragma unroll
        for (int j = 0; j < 16; ++j) {
            int m = tm + 8 * j;
            async_copy_b32(lds_addr(&As[panel][m * 16 + rr]), orow + m);
            async_copy_b32(lds_addr(&As[panel][(Mn + m) * 16 + rr]), irow + m);
        }
        S_WAIT_ASYNC_0();
        __syncthreads();
    }

    v8f acc0 = {}, acc1 = {};
    for (int k0 = 0; k0 < 2 * Mn; k0 += 4) {
        v2f a0, a1;
        a0.x = As[0][(k0 + kb) * 16 + rn];
        a0.y = As[0][(k0 + kb + 1) * 16 + rn];
        a1.x = As[1][(k0 + kb) * 16 + rn];
        a1.y = As[1][(k0 + kb + 1) * 16 + rn];
        v2f bt;
        bt.x = twI[(size_t)(k0 + kb) * Ln + l0 + rn];
        bt.y = twI[(size_t)(k0 + kb + 1) * Ln + l0 + rn];
        acc0 = wmma_f32(a0, bt, acc0);
        acc1 = wmma_f32(a1, bt, acc1);
    }

#pragma unroll
    for (int v = 0; v < 8; ++v) {
        int er = v + hi * 8;
        out[(size_t)(row0 + er) * Ln + (l0 + rn)]      = acc0[v];
        out[(size_t)(row0 + 16 + er) * Ln + (l0 + rn)] = acc1[v];
    }
}

// ---------------------------------------------------------------------------
extern "C" void kernel_launch(void* const* d_in, const int* in_sizes, int n_in,
                              void* d_out, int out_size, void* d_ws, size_t ws_size,
                              hipStream_t stream) {
    const float* q      = (const float*)d_in[0];   // [B,L,H,E]
    const float* w_real = (const float*)d_in[1];   // [H,E,E,M]
    const float* w_imag = (const float*)d_in[2];   // [H,E,E,M]
    const int*   index  = (const int*)d_in[3];     // [M]
    (void)in_sizes; (void)n_in;                    // mask (d_in[4]) unused
    float* out = (float*)d_out;                    // [B,H,E,L]

    // workspace carve-up (floats)
    float* twF = (float*)d_ws;                          // L * 2M   = 1,048,576
    float* twI = twF + (size_t)Ln * 2 * Mn;             // 2M * L   = 1,048,576
    float* Xr2 = twI + (size_t)2 * Mn * Ln;             // H*M*B*E  = 3,145,728
    float* Xi2 = Xr2 + (size_t)Hn * Mn * Bn * En;
    float* Or  = Xi2 + (size_t)Hn * Mn * Bn * En;       // B*H*E*M  = 3,145,728
    float* Oi  = Or  + (size_t)Bn * Hn * En * Mn;
    (void)ws_size;

    // K0: twiddles (M*L threads)
    fb_twiddles<<<(Mn * Ln) / 256, 256, 0, stream>>>(index, twF, twI);

    // K1: forward DFT — R/32 = 768 row tiles (32 rows x 128 modes each)
    fb_dft_fwd<<<Rn / 32, 256, 0, stream>>>(q, twF, Xr2, Xi2);

    // K2: mode mixing — 384 (h,m) pairs x 2 o-halves; 2 o-tiles per wave
    fb_modemix<<<Hn * Mn * 2, 256, 0, stream>>>(Xr2, Xi2, w_real, w_imag, Or, Oi);

    // K3: inverse DFT — 768 row tiles x 32 l-chunks (8 l-tiles per block)
    fb_idft<<<dim3(Rn / 32, Ln / 128), 256, 0, stream>>>(Or, Oi, twI, out);

    (void)out_size; (void)out;
}